// MdpTorchKernel_60318520705713
// MI455X (gfx1250) — compile-verified
//
#include <hip/hip_runtime.h>
#include <hip/hip_bf16.h>
#include <math.h>

// Dims (fixed by the reference)
#define T_ 64
#define S_ 512
#define X_ 16
#define Y_ 16
#define G_ 64

typedef __attribute__((ext_vector_type(2))) float v2f;
typedef __attribute__((ext_vector_type(8))) float v8f;

// J = sum_s pi_0[s] * V[0,s], with
// V[0,s] = sum_{x,y,g} px0[x]*(loss[s,x,y]+x+y)*py0[y,g]*obv[s,g]
// (Phi term of the reference is identically zero: V_pre is zeros.)
__global__ __launch_bounds__(1024)
void mdp_j_kernel(const float* __restrict__ policy_var,
                  const float* __restrict__ obv,        // [S,G]
                  const float* __restrict__ loss_table, // [S,X,Y]
                  const float* __restrict__ pi_0,       // [S]
                  float* __restrict__ out)              // [1]
{
    __shared__ __align__(16) float pxLds[X_];
    __shared__ float cxbarLds;
    __shared__ __align__(16) float pyT[Y_][G_];        // pyT[y][g] = py0[y,g]
    __shared__ __align__(16) float Bmat[S_][Y_ + 1];   // +1 pad: conflict-free col reads
    __shared__ __align__(16) float red[S_];

    const int tid = threadIdx.x;

    // ---------- Phase A: softmaxes (t = 0 rows only) ----------
    if (tid == 0) {
        float v[X_], mx = -3.4e38f;
        #pragma unroll
        for (int x = 0; x < X_; ++x) { v[x] = policy_var[x]; mx = fmaxf(mx, v[x]); }
        float ssum = 0.f;
        #pragma unroll
        for (int x = 0; x < X_; ++x) { v[x] = expf(v[x] - mx); ssum += v[x]; }
        float inv = 1.f / ssum, cxb = 0.f;
        #pragma unroll
        for (int x = 0; x < X_; ++x) {
            float p = v[x] * inv;
            pxLds[x] = p;
            cxb += p * (float)x;           // sum_x px0[x]*x  (c_x = LAM = 1)
        }
        cxbarLds = cxb;
    }
    if (tid < G_) {                        // one thread per g column, softmax over y
        const int g = tid;
        float v[Y_], mx = -3.4e38f;
        #pragma unroll
        for (int y = 0; y < Y_; ++y) {
            v[y] = policy_var[T_ * X_ + y * G_ + g];   // t = 0
            mx = fmaxf(mx, v[y]);
        }
        float ssum = 0.f;
        #pragma unroll
        for (int y = 0; y < Y_; ++y) { v[y] = expf(v[y] - mx); ssum += v[y]; }
        float inv = 1.f / ssum;
        #pragma unroll
        for (int y = 0; y < Y_; ++y) pyT[y][g] = v[y] * inv;
    }
    __syncthreads();

    // ---------- Phase B: Bmat[s][y] = sum_g obv[s,g]*py0[y,g] ----------
    // [512,64] x [64,16] GEMM via V_WMMA_F32_16X16X4_F32; wave w owns s-tile w.
    // No divergence here: EXEC must be all 1s for WMMA.
    {
        const int wave = tid >> 5;         // 0..31  -> s-tile
        const int lane = tid & 31;
        const int h    = lane >> 4;        // K half-select (A & B) / M half (D)
        const int n    = lane & 15;        // A row (M) and B col (N)
        const float* arow = obv + (size_t)(wave * 16 + n) * G_;
        v8f acc = {0.f, 0.f, 0.f, 0.f, 0.f, 0.f, 0.f, 0.f};
        #pragma unroll
        for (int kk = 0; kk < 16; ++kk) {  // 16 chained K=4 steps over g
            const int kb = kk * 4 + h * 2; // this lane's K base (2 consecutive g)
            v2f a = *reinterpret_cast<const v2f*>(arow + kb);        // A[m,kb..kb+1]
            v2f b = *reinterpret_cast<const v2f*>(&pyT[n][kb]);      // B[kb..kb+1,n]
            acc = __builtin_amdgcn_wmma_f32_16x16x4_f32(
                      /*neg_a=*/false, a, /*neg_b=*/false, b,
                      /*c_mod=*/(short)0, acc, /*reuse_a=*/false, /*reuse_b=*/false);
        }
        // D layout: lane l, VGPR v -> (M = v + 8*(l/16), N = l&15)
        #pragma unroll
        for (int v = 0; v < 8; ++v)
            Bmat[wave * 16 + h * 8 + v][n] = acc[v];
    }
    __syncthreads();

    // ---------- Phase C: per-s contraction + pi_0 dot ----------
    if (tid < S_) {
        const int s = tid;
        const float* Ls  = loss_table + (size_t)s * (X_ * Y_);
        const float cxb  = cxbarLds;
        float V0 = 0.f;
        #pragma unroll
        for (int y = 0; y < Y_; ++y) {
            float d = 0.f;
            #pragma unroll
            for (int x = 0; x < X_; ++x)
                d = fmaf(pxLds[x], Ls[x * Y_ + y], d);
            V0 = fmaf(d + cxb + (float)y, Bmat[s][y], V0);
        }
        red[s] = pi_0[s] * V0;
    }
    __syncthreads();

    // ---------- Tree reduction over s ----------
    #pragma unroll
    for (int off = 256; off > 0; off >>= 1) {
        if (tid < off) red[tid] += red[tid + off];
        __syncthreads();
    }
    if (tid == 0) out[0] = red[0];
}

extern "C" void kernel_launch(void* const* d_in, const int* in_sizes, int n_in,
                              void* d_out, int out_size, void* d_ws, size_t ws_size,
                              hipStream_t stream) {
    const float* policy_var = (const float*)d_in[0];
    // d_in[1] = trans: multiplied by an all-zero V_pre in the reference -> never read.
    const float* obv        = (const float*)d_in[2];
    const float* loss_table = (const float*)d_in[3];
    const float* pi_0       = (const float*)d_in[4];
    float* out = (float*)d_out;
    (void)in_sizes; (void)n_in; (void)out_size; (void)d_ws; (void)ws_size;

    mdp_j_kernel<<<1, 1024, 0, stream>>>(policy_var, obv, loss_table, pi_0, out);
}